// EvolveGCN_H_7327214207525
// MI455X (gfx1250) — compile-verified
//
#include <hip/hip_runtime.h>
#include <hip/hip_bf16.h>
#include <math.h>

// ---------------------------------------------------------------------------
// EvolveGCN-H style pipeline for MI455X (gfx1250, wave32, WMMA).
// fp32 throughout (pipeline is HBM/L2-bandwidth bound, not FLOP bound).
// ---------------------------------------------------------------------------

typedef __attribute__((ext_vector_type(2))) float v2f;
typedef __attribute__((ext_vector_type(8))) float v8f;

#define D 128                 // feature dim / hidden dim / k of top-k
#define PAIR_STRIDE 288       // dwords per K-pair row of B in LDS:
                              // 288 % 64 == 32 -> lanes 16..31 (next pair row)
                              // use the other 32 banks -> conflict-free b64 reads

// ---------------------------------------------------------------------------
// 1) score[i] = dot(x[i,:], p) / ||p||        (one wave32 per node)
// ---------------------------------------------------------------------------
__global__ void score_kernel(const float* __restrict__ x,
                             const float* __restrict__ p,
                             float* __restrict__ scores, int N) {
    const int lane = threadIdx.x & 31;
    const int wid  = (blockIdx.x * blockDim.x + threadIdx.x) >> 5;
    const int nw   = (gridDim.x * blockDim.x) >> 5;

    float4 pv = ((const float4*)p)[lane];            // p[4*lane .. 4*lane+3]
    float pn = pv.x * pv.x + pv.y * pv.y + pv.z * pv.z + pv.w * pv.w;
    #pragma unroll
    for (int o = 16; o > 0; o >>= 1) pn += __shfl_xor(pn, o, 32);
    const float inv_norm = rsqrtf(pn);

    for (int i = wid; i < N; i += nw) {
        float4 xv = ((const float4*)(x + (size_t)i * D))[lane];
        float s = xv.x * pv.x + xv.y * pv.y + xv.z * pv.z + xv.w * pv.w;
        #pragma unroll
        for (int o = 16; o > 0; o >>= 1) s += __shfl_xor(s, o, 32);
        if (lane == 0) scores[i] = s * inv_norm;
    }
}

// ---------------------------------------------------------------------------
// 2) top-128 selection: 128 iterative argmax passes (ties -> lowest index,
//    matching jax.lax.top_k). Destroys scores (recomputed every call).
// ---------------------------------------------------------------------------
__global__ void topk_kernel(float* __restrict__ scores, int N,
                            int* __restrict__ perm, float* __restrict__ topv) {
    __shared__ float sv[1024];
    __shared__ int   si[1024];
    const int tid = threadIdx.x;
    for (int t = 0; t < D; ++t) {
        float bv = -INFINITY; int bi = 0x7fffffff;
        for (int i = tid; i < N; i += 1024) {
            float v = scores[i];
            if (v > bv || (v == bv && i < bi)) { bv = v; bi = i; }
        }
        sv[tid] = bv; si[tid] = bi;
        __syncthreads();
        for (int s = 512; s > 0; s >>= 1) {
            if (tid < s) {
                float ov = sv[tid + s]; int oi = si[tid + s];
                if (ov > sv[tid] || (ov == sv[tid] && oi < si[tid])) {
                    sv[tid] = ov; si[tid] = oi;
                }
            }
            __syncthreads();
        }
        if (tid == 0) {
            perm[t] = si[0];
            topv[t] = sv[0];
            scores[si[0]] = -INFINITY;   // remove from next pass
        }
        __syncthreads();
    }
}

// ---------------------------------------------------------------------------
// 3) x_tilde[r,c] = x[perm[r],c] * tanh(topv[r])
// ---------------------------------------------------------------------------
__global__ void xtilde_kernel(const float* __restrict__ x,
                              const int* __restrict__ perm,
                              const float* __restrict__ topv,
                              float* __restrict__ xt) {
    const int r = blockIdx.x, c = threadIdx.x;
    xt[r * D + c] = x[(size_t)perm[r] * D + c] * tanhf(topv[r]);
}

// ---------------------------------------------------------------------------
// 4) GRU single step: h0 = W0 rows, batch = 128 rows of x_tilde.
//    One block per row i; thread j computes output column j.
// ---------------------------------------------------------------------------
__global__ void gru_kernel(const float* __restrict__ xt,
                           const float* __restrict__ W0,
                           const float* __restrict__ W_ih,
                           const float* __restrict__ W_hh,
                           const float* __restrict__ b_ih,
                           const float* __restrict__ b_hh,
                           float* __restrict__ Wout) {
    const int i = blockIdx.x;    // row 0..127
    const int j = threadIdx.x;   // col 0..127
    __shared__ float xrow[D];
    __shared__ float hrow[D];
    xrow[j] = xt[i * D + j];
    hrow[j] = W0[i * D + j];
    __syncthreads();

    float gxr = b_ih[j], gxz = b_ih[j + D], gxn = b_ih[j + 2 * D];
    float ghr = b_hh[j], ghz = b_hh[j + D], ghn = b_hh[j + 2 * D];
    const float* wi0 = W_ih + (size_t)(j) * D;
    const float* wi1 = W_ih + (size_t)(j + D) * D;
    const float* wi2 = W_ih + (size_t)(j + 2 * D) * D;
    const float* wh0 = W_hh + (size_t)(j) * D;
    const float* wh1 = W_hh + (size_t)(j + D) * D;
    const float* wh2 = W_hh + (size_t)(j + 2 * D) * D;
    #pragma unroll 4
    for (int k = 0; k < D; ++k) {
        const float xv = xrow[k], hv = hrow[k];
        gxr += xv * wi0[k]; gxz += xv * wi1[k]; gxn += xv * wi2[k];
        ghr += hv * wh0[k]; ghz += hv * wh1[k]; ghn += hv * wh2[k];
    }
    const float r = 1.0f / (1.0f + __expf(-(gxr + ghr)));
    const float z = 1.0f / (1.0f + __expf(-(gxz + ghz)));
    const float n = tanhf(gxn + r * ghn);
    Wout[i * D + j] = (1.0f - z) * n + z * hrow[j];
}

// ---------------------------------------------------------------------------
// 5) degree (self-loop folded in as init=1), then in-place rsqrt
// ---------------------------------------------------------------------------
__global__ void deg_init(float* __restrict__ deg, int N) {
    int i = blockIdx.x * blockDim.x + threadIdx.x;
    if (i < N) deg[i] = 1.0f;                // self loop
}
__global__ void deg_count(const int* __restrict__ ei, float* __restrict__ deg, int E) {
    int e = blockIdx.x * blockDim.x + threadIdx.x;
    if (e < E) atomicAdd(&deg[ei[E + e]], 1.0f);   // dst row of edge_index
}
__global__ void deg_rsqrt(float* __restrict__ deg, int N) {
    int i = blockIdx.x * blockDim.x + threadIdx.x;
    if (i < N) deg[i] = rsqrtf(deg[i]);      // deg >= 1 always
}

// ---------------------------------------------------------------------------
// 6) WMMA fp32 GEMM: C[M x 128] = f(A[M x 128]) * B' (+ bias)
//    B' = B (128x128) or B^T when TRANS_B. RELU_A applies relu to A on load.
//    Block = 256 threads = 8 waves; each wave owns a 16-row strip and all
//    8 column tiles (64 accum VGPRs). B staged in LDS *pair-interleaved*:
//      Bs[(k>>1)*PAIR_STRIDE + 2n + (k&1)]
//    so a B fragment (needs B[k][n], B[k+1][n] in one even VGPR pair) is a
//    single aligned ds_load_b64, conflict-free across the wave.
//    V_WMMA_F32_16X16X4_F32 fragment striping (CDNA5 ISA):
//      A (16x4): lane l -> row (l&15), VGPR v -> K = 2*(l>>4)+v
//      B (4x16): lane l -> col (l&15), VGPR v -> K = 2*(l>>4)+v
//      C (16x16): VGPR j, lane l -> row j+8*(l>>4), col (l&15)
// ---------------------------------------------------------------------------
template <bool RELU_A, bool TRANS_B, bool BIAS>
__global__ void gemm128_wmma(const float* __restrict__ A,
                             const float* __restrict__ B,
                             const float* __restrict__ bias,
                             float* __restrict__ C, int M) {
    __shared__ float Bs[(D / 2) * PAIR_STRIDE];   // 64 pair-rows, 73.7 KB

    const int tid  = threadIdx.x;
    const int wave = tid >> 5;
    const int lane = tid & 31;

    // cooperative B load (64 KB matrix, once per block), pair-interleaved
    for (int i = tid; i < D * D; i += 256) {
        const int k = i >> 7, n = i & (D - 1);
        const float v = TRANS_B ? B[n * D + k] : B[i];
        Bs[(k >> 1) * PAIR_STRIDE + 2 * n + (k & 1)] = v;
    }
    __syncthreads();

    const int m0    = blockIdx.x * 128 + wave * 16;   // this wave's row strip
    const int halfr = lane >> 4;                      // 0 | 1
    const int lr    = lane & 15;
    const int m     = m0 + lr;
    const int mc    = m < M ? m : (M - 1);            // clamp for safe loads
    const int khalf = halfr * 2;                      // K sub-offset 0 | 2

    v8f acc[8] = {};

    for (int k0 = 0; k0 < D; k0 += 4) {
        v2f a = *(const v2f*)(A + (size_t)mc * D + k0 + khalf);
        if (RELU_A) { a.x = fmaxf(a.x, 0.0f); a.y = fmaxf(a.y, 0.0f); }
        // pair-row for this lane half: (k0 + khalf) / 2
        const float* bp = &Bs[((k0 >> 1) + halfr) * PAIR_STRIDE];
        #pragma unroll
        for (int t = 0; t < 8; ++t) {
            const v2f b = *(const v2f*)(bp + 2 * (t * 16 + lr));  // ds_load_b64
            acc[t] = __builtin_amdgcn_wmma_f32_16x16x4_f32(
                false, a, false, b, (short)0, acc[t], false, false);
        }
    }

    const int rowoff = halfr * 8;
    #pragma unroll
    for (int t = 0; t < 8; ++t) {
        const int n0 = t * 16;
        const float bv = BIAS ? bias[n0 + lr] : 0.0f;
        #pragma unroll
        for (int j = 0; j < 8; ++j) {
            const int r = m0 + rowoff + j;
            if (r < M) C[(size_t)r * D + n0 + lr] = acc[t][j] + bv;
        }
    }
}

// ---------------------------------------------------------------------------
// 7) agg init: self-loop message agg[i,:] = h[i,:] * dis[i]^2 (also clears)
// ---------------------------------------------------------------------------
__global__ void agg_init(const float* __restrict__ h,
                         const float* __restrict__ dis,
                         float* __restrict__ agg, int N) {
    size_t idx = (size_t)blockIdx.x * blockDim.x + threadIdx.x;
    if (idx < (size_t)N * D) {
        const int i = (int)(idx >> 7);
        const float d = dis[i];
        agg[idx] = h[idx] * d * d;
    }
}

// ---------------------------------------------------------------------------
// 8) edge scatter: one wave per edge, float4 gather, 4 atomics per lane.
//    agg[dst,:] += h[src,:] * dis[src]*dis[dst]   (atomics resolve in L2;
//    agg is 51 MB, fully L2-resident on MI455X's 192 MB L2)
// ---------------------------------------------------------------------------
__global__ void scatter_kernel(const float* __restrict__ h,
                               const float* __restrict__ dis,
                               const int* __restrict__ ei,
                               float* __restrict__ agg, int E) {
    const size_t gid = (size_t)blockIdx.x * blockDim.x + threadIdx.x;
    const int e = (int)(gid >> 5);
    if (e >= E) return;
    const int lane = (int)(gid & 31);
    const int s = ei[e];
    const int d = ei[E + e];
    const float w = dis[s] * dis[d];
    float4 v = ((const float4*)(h + (size_t)s * D))[lane];
    float* ap = agg + (size_t)d * D + lane * 4;
    atomicAdd(ap + 0, v.x * w);
    atomicAdd(ap + 1, v.y * w);
    atomicAdd(ap + 2, v.z * w);
    atomicAdd(ap + 3, v.w * w);
}

// ---------------------------------------------------------------------------
// launch
// ---------------------------------------------------------------------------
extern "C" void kernel_launch(void* const* d_in, const int* in_sizes, int n_in,
                              void* d_out, int out_size, void* d_ws, size_t ws_size,
                              hipStream_t stream) {
    const float* x       = (const float*)d_in[0];
    const int*   ei      = (const int*)d_in[1];
    const float* pool_p  = (const float*)d_in[2];
    const float* W_ih    = (const float*)d_in[3];
    const float* W_hh    = (const float*)d_in[4];
    const float* b_ih    = (const float*)d_in[5];
    const float* b_hh    = (const float*)d_in[6];
    const float* W0      = (const float*)d_in[7];
    const float* lin_W   = (const float*)d_in[8];
    const float* lin_b   = (const float*)d_in[9];
    float*       out     = (float*)d_out;

    const int N = in_sizes[0] / D;       // 100000
    const int E = in_sizes[1] / 2;       // 1600000

    // workspace layout
    float* h      = (float*)d_ws;                 // N*128
    float* scores = h + (size_t)N * D;            // N
    float* dis    = scores + N;                   // N (deg -> rsqrt in place)
    float* xt     = dis + N;                      // 128*128
    float* Wevo   = xt + D * D;                   // 128*128
    float* topv   = Wevo + D * D;                 // 128
    int*   perm   = (int*)(topv + D);             // 128

    // --- TopKPooling ---
    score_kernel<<<1024, 256, 0, stream>>>(x, pool_p, scores, N);
    topk_kernel<<<1, 1024, 0, stream>>>(scores, N, perm, topv);
    xtilde_kernel<<<D, D, 0, stream>>>(x, perm, topv, xt);

    // --- GRU evolves the GCN weight ---
    gru_kernel<<<D, D, 0, stream>>>(xt, W0, W_ih, W_hh, b_ih, b_hh, Wevo);

    // --- degrees / symmetric norm ---
    deg_init<<<(N + 255) / 256, 256, 0, stream>>>(dis, N);
    deg_count<<<(E + 255) / 256, 256, 0, stream>>>(ei, dis, E);
    deg_rsqrt<<<(N + 255) / 256, 256, 0, stream>>>(dis, N);

    // --- h = x @ Wevo (WMMA fp32) ---
    gemm128_wmma<false, false, false><<<(N + 127) / 128, 256, 0, stream>>>(
        x, Wevo, nullptr, h, N);

    // --- aggregate: self loops then edges (into d_out) ---
    agg_init<<<(int)(((size_t)N * D + 255) / 256), 256, 0, stream>>>(h, dis, out, N);
    scatter_kernel<<<(int)(((size_t)E * 32 + 255) / 256), 256, 0, stream>>>(
        h, dis, ei, out, E);

    // --- out = relu(agg) @ lin_W^T + lin_b, in place on d_out ---
    gemm128_wmma<true, true, true><<<(N + 127) / 128, 256, 0, stream>>>(
        out, lin_W, lin_b, out, N);
}